// KalmanFilter_74105365725206
// MI455X (gfx1250) — compile-verified
//
// Kalman filter update, batched B*N = 524288 problems (STATE=8, OBS=4), f32.
// MI455X (gfx1250, wave32). Strategy:
//   - 4 problems packed per 16x16 WMMA tile (quadrants); blkdiag(F,F) (x) tile (x) blkdiag(F^T,F^T)
//     computes F P F^T for all 4 quadrants with V_WMMA_F32_16X16X4_F32 chains (K accumulation).
//   - P' H^T also via WMMA with blkdiag([H^T|0]).
//   - Per-thread VALU: innovation, S = H P' H^T + R, explicit 4x4 inverse, K, state update,
//     cov_new = P' - K (P'H^T)^T.
//   - LDS staging for WMMA layout shuffles + coalesced cov_new store. No inter-stage barriers
//     needed: each wave's threads consume the problems that same wave produced (DS in-order).
#include <hip/hip_runtime.h>

#define WG_THREADS 128
#define PROB_PER_WG 128
#define TILES_PER_WAVE 8   // 32 problems per wave / 4 per tile

typedef float v2f __attribute__((ext_vector_type(2)));
typedef float v8f __attribute__((ext_vector_type(8)));

#define COV_STR 68
#define PHT_STR 36
#define T1_STR  17

__global__ __launch_bounds__(WG_THREADS)
void kalman_wmma_kernel(const float* __restrict__ g_state,
                        const float* __restrict__ g_cov,
                        const float* __restrict__ g_obs,
                        const float* __restrict__ g_F,
                        const float* __restrict__ g_H,
                        const float* __restrict__ g_Q,
                        const float* __restrict__ g_R,
                        float* __restrict__ out_state,
                        float* __restrict__ out_cov)
{
    __shared__ float sF[64], sH[32], sQ[64], sR[16];
    __shared__ float sT1[4][16 * T1_STR];
    __shared__ float sCov[PROB_PER_WG * COV_STR];
    __shared__ float sPht[PROB_PER_WG * PHT_STR];

    const int tid  = threadIdx.x;
    const int wave = tid >> 5;
    const int lane = tid & 31;
    const int wg_p0 = blockIdx.x * PROB_PER_WG;

    if (tid < 64) { sF[tid] = g_F[tid]; sQ[tid] = g_Q[tid]; }
    if (tid < 32) sH[tid] = g_H[tid];
    if (tid < 16) sR[tid] = g_R[tid];
    __syncthreads();

    const int n16  = lane & 15;
    const int half = lane >> 4;

    // Per-lane constant operand slices.
    // fa[k]: A-layout slice k of blkdiag(F,F): element (m=n16, c=4k+2*half+j)
    // fb[k]: B-layout slice k of blkdiag(F^T,F^T): element (r=4k+2*half+j, n=n16)
    // hb[k]: B-layout slice k of blkdiag([H^T|0],[H^T|0])
    v2f fa[4], fb[4], hb[4];
#pragma unroll
    for (int k = 0; k < 4; ++k) {
#pragma unroll
        for (int j = 0; j < 2; ++j) {
            const int c = 4 * k + 2 * half + j;
            const bool sameBlkA = ((n16 < 8) == (c < 8));
            fa[k][j] = sameBlkA ? sF[(n16 & 7) * 8 + (c & 7)] : 0.0f;
            fb[k][j] = sameBlkA ? sF[(n16 & 7) * 8 + (c & 7)] : 0.0f;  // F^T[r][n] = F[n][r]
            hb[k][j] = (sameBlkA && ((n16 & 7) < 4)) ? sH[(n16 & 7) * 8 + (c & 7)] : 0.0f;
        }
    }

    const int wave_p0 = wg_p0 + 32 * wave;
    const int lbase0  = 32 * wave;
    float* t1 = &sT1[wave][0];

    for (int t = 0; t < TILES_PER_WAVE; ++t) {
        const int p0 = wave_p0 + 4 * t;   // problem of quadrant (0,0)
        const int l0 = lbase0 + 4 * t;

        // Prefetch next tile's covariances (speculative; dropped if OOB).
        __builtin_prefetch(&g_cov[(size_t)(p0 + 4) * 64], 0, 1);

        // ---- Pass A: T1 = blkdiag(F,F) x Ptile  (quadrant(qm,qn) = P_{2qm+qn})
        v8f acc = {};
#pragma unroll
        for (int k = 0; k < 4; ++k) {
            v2f b;
#pragma unroll
            for (int j = 0; j < 2; ++j) {
                const int r = 4 * k + 2 * half + j;
                const int o = 2 * (r >> 3) + (n16 >> 3);
                b[j] = g_cov[(size_t)(p0 + o) * 64 + (r & 7) * 8 + (n16 & 7)];
            }
            acc = __builtin_amdgcn_wmma_f32_16x16x4_f32(false, fa[k], false, b,
                                                        (short)0, acc, false, false);
        }
#pragma unroll
        for (int v = 0; v < 8; ++v)
            t1[(v + 8 * half) * T1_STR + n16] = acc[v];

        // ---- Pass B: cov_pred = T1 x blkdiag(F^T,F^T) + Q (per quadrant)
        v8f cc;
#pragma unroll
        for (int v = 0; v < 8; ++v) cc[v] = sQ[v * 8 + (n16 & 7)];
#pragma unroll
        for (int k = 0; k < 4; ++k) {
            v2f a;
            a[0] = t1[n16 * T1_STR + 4 * k + 2 * half + 0];
            a[1] = t1[n16 * T1_STR + 4 * k + 2 * half + 1];
            cc = __builtin_amdgcn_wmma_f32_16x16x4_f32(false, a, false, fb[k],
                                                       (short)0, cc, false, false);
        }
        {
            const int o = 2 * half + (n16 >> 3);
            float* cdst = &sCov[(l0 + o) * COV_STR];
#pragma unroll
            for (int v = 0; v < 8; ++v)
                cdst[v * 8 + (n16 & 7)] = cc[v];
        }

        // ---- Pass C: PHT = cov_pred_tile x blkdiag([H^T|0]) (valid in quadrant cols 0..3)
        v8f pc = {};
#pragma unroll
        for (int k = 0; k < 4; ++k) {
            v2f a;
#pragma unroll
            for (int j = 0; j < 2; ++j) {
                const int c = 4 * k + 2 * half + j;
                const int o = 2 * (n16 >> 3) + (c >> 3);
                a[j] = sCov[(l0 + o) * COV_STR + (n16 & 7) * 8 + (c & 7)];
            }
            pc = __builtin_amdgcn_wmma_f32_16x16x4_f32(false, a, false, hb[k],
                                                       (short)0, pc, false, false);
        }
        if ((n16 & 7) < 4) {
            const int o = 2 * half + (n16 >> 3);
            float* pdst = &sPht[(l0 + o) * PHT_STR];
#pragma unroll
            for (int v = 0; v < 8; ++v)
                pdst[v * 4 + (n16 & 7)] = pc[v];
        }
    }

    // ---- Per-thread update. Thread tid owns local problem tid (produced by its own wave,
    // so intra-wave LDS ordering suffices; no barrier needed here).
    const size_t p = (size_t)wg_p0 + tid;
    float s[8], ob[4], sp[8], y[4];
#pragma unroll
    for (int i = 0; i < 8; ++i) s[i] = g_state[p * 8 + i];
#pragma unroll
    for (int a = 0; a < 4; ++a) ob[a] = g_obs[p * 4 + a];
#pragma unroll
    for (int i = 0; i < 8; ++i) {
        float acc = 0.0f;
#pragma unroll
        for (int j = 0; j < 8; ++j) acc += sF[i * 8 + j] * s[j];
        sp[i] = acc;
    }
#pragma unroll
    for (int a = 0; a < 4; ++a) {
        float acc = ob[a];
#pragma unroll
        for (int j = 0; j < 8; ++j) acc -= sH[a * 8 + j] * sp[j];
        y[a] = acc;
    }

    float pht[8][4];
    {
        const float* psrc = &sPht[tid * PHT_STR];
#pragma unroll
        for (int i = 0; i < 8; ++i)
#pragma unroll
            for (int a = 0; a < 4; ++a) pht[i][a] = psrc[i * 4 + a];
    }

    // S = H * PHT + R  (4x4)
    float S[4][4];
#pragma unroll
    for (int a = 0; a < 4; ++a)
#pragma unroll
        for (int b = 0; b < 4; ++b) {
            float acc = sR[a * 4 + b];
#pragma unroll
            for (int i = 0; i < 8; ++i) acc += sH[a * 8 + i] * pht[i][b];
            S[a][b] = acc;
        }

    // Explicit 4x4 inverse (cofactors / 2x2 subdeterminants).
    float Si[4][4];
    {
        const float a00 = S[0][0], a01 = S[0][1], a02 = S[0][2], a03 = S[0][3];
        const float a10 = S[1][0], a11 = S[1][1], a12 = S[1][2], a13 = S[1][3];
        const float a20 = S[2][0], a21 = S[2][1], a22 = S[2][2], a23 = S[2][3];
        const float a30 = S[3][0], a31 = S[3][1], a32 = S[3][2], a33 = S[3][3];
        const float s0 = a00 * a11 - a10 * a01;
        const float s1 = a00 * a12 - a10 * a02;
        const float s2 = a00 * a13 - a10 * a03;
        const float s3 = a01 * a12 - a11 * a02;
        const float s4 = a01 * a13 - a11 * a03;
        const float s5 = a02 * a13 - a12 * a03;
        const float c5 = a22 * a33 - a32 * a23;
        const float c4 = a21 * a33 - a31 * a23;
        const float c3 = a21 * a32 - a31 * a22;
        const float c2 = a20 * a33 - a30 * a23;
        const float c1 = a20 * a32 - a30 * a22;
        const float c0 = a20 * a31 - a30 * a21;
        const float id = 1.0f / (s0 * c5 - s1 * c4 + s2 * c3 + s3 * c2 - s4 * c1 + s5 * c0);
        Si[0][0] = ( a11 * c5 - a12 * c4 + a13 * c3) * id;
        Si[0][1] = (-a01 * c5 + a02 * c4 - a03 * c3) * id;
        Si[0][2] = ( a31 * s5 - a32 * s4 + a33 * s3) * id;
        Si[0][3] = (-a21 * s5 + a22 * s4 - a23 * s3) * id;
        Si[1][0] = (-a10 * c5 + a12 * c2 - a13 * c1) * id;
        Si[1][1] = ( a00 * c5 - a02 * c2 + a03 * c1) * id;
        Si[1][2] = (-a30 * s5 + a32 * s2 - a33 * s1) * id;
        Si[1][3] = ( a20 * s5 - a22 * s2 + a23 * s1) * id;
        Si[2][0] = ( a10 * c4 - a11 * c2 + a13 * c0) * id;
        Si[2][1] = (-a00 * c4 + a01 * c2 - a03 * c0) * id;
        Si[2][2] = ( a30 * s4 - a31 * s2 + a33 * s0) * id;
        Si[2][3] = (-a20 * s4 + a21 * s2 - a23 * s0) * id;
        Si[3][0] = (-a10 * c3 + a11 * c1 - a12 * c0) * id;
        Si[3][1] = ( a00 * c3 - a01 * c1 + a02 * c0) * id;
        Si[3][2] = (-a30 * s3 + a31 * s1 - a32 * s0) * id;
        Si[3][3] = ( a20 * s3 - a21 * s1 + a22 * s0) * id;
    }

    // K = PHT * Sinv (8x4)
    float K[8][4];
#pragma unroll
    for (int i = 0; i < 8; ++i)
#pragma unroll
        for (int b = 0; b < 4; ++b) {
            float acc = 0.0f;
#pragma unroll
            for (int a = 0; a < 4; ++a) acc += pht[i][a] * Si[a][b];
            K[i][b] = acc;
        }

    // state_new = state_pred + K y
#pragma unroll
    for (int i = 0; i < 8; ++i) {
        float acc = sp[i];
#pragma unroll
        for (int a = 0; a < 4; ++a) acc += K[i][a] * y[a];
        out_state[p * 8 + i] = acc;
    }

    // cov_new = cov_pred - K * PHT^T  (in place in LDS), then coalesced store.
    {
        float* crow = &sCov[tid * COV_STR];
#pragma unroll
        for (int i = 0; i < 8; ++i)
#pragma unroll
            for (int j = 0; j < 8; ++j) {
                float acc = crow[i * 8 + j];
#pragma unroll
                for (int a = 0; a < 4; ++a) acc -= K[i][a] * pht[j][a];
                crow[i * 8 + j] = acc;
            }
    }
    __syncthreads();
    {
        const size_t base = (size_t)wg_p0 * 64;
        for (int idx = tid; idx < PROB_PER_WG * 64; idx += WG_THREADS) {
            const int pl = idx >> 6, c = idx & 63;
            out_cov[base + idx] = sCov[pl * COV_STR + c];
        }
    }
}

extern "C" void kernel_launch(void* const* d_in, const int* in_sizes, int n_in,
                              void* d_out, int out_size, void* d_ws, size_t ws_size,
                              hipStream_t stream) {
    const float* g_state = (const float*)d_in[0];
    const float* g_cov   = (const float*)d_in[1];
    const float* g_obs   = (const float*)d_in[2];
    const float* g_F     = (const float*)d_in[3];
    const float* g_H     = (const float*)d_in[4];
    const float* g_Q     = (const float*)d_in[5];
    const float* g_R     = (const float*)d_in[6];

    const int n_prob = in_sizes[0] / 8;            // state is (B,N,8); 128*4096 = 524288
    float* out_state = (float*)d_out;
    float* out_cov   = out_state + (size_t)n_prob * 8;

    const int n_wg = n_prob / PROB_PER_WG;         // 4096 (n_prob divisible by 128)
    kalman_wmma_kernel<<<n_wg, WG_THREADS, 0, stream>>>(
        g_state, g_cov, g_obs, g_F, g_H, g_Q, g_R, out_state, out_cov);
}